// Aggregator_90890097918063
// MI455X (gfx1250) — compile-verified
//
#include <hip/hip_runtime.h>

// ---------------------------------------------------------------------------
// KGAT-style aggregator for MI455X (gfx1250, wave32).
// Memory-bound scatter/gather done with float4 loads + f32 global atomics;
// the dense user-side GEMMs + softmax run on V_WMMA_F32_16X16X4_F32.
// ---------------------------------------------------------------------------

#define D128 128

typedef __attribute__((ext_vector_type(2))) float v2f;
typedef __attribute__((ext_vector_type(8))) float v8f;

// ---------------------------------------------------------------- zero fill
__global__ void zero_kernel(float* __restrict__ p, long long n) {
  long long i = (long long)blockIdx.x * blockDim.x + threadIdx.x;
  long long st = (long long)gridDim.x * blockDim.x;
  for (; i < n; i += st) p[i] = 0.0f;
}

// ------------------------------------------------------- pass 1: att + seg-max
// One wave per edge. att = ||h*r||^2 * ||t*r||^2  (== (||h*r||*||t*r||)^2).
// Segment max via atomicMax on uint bit pattern (att >= 0 so ordering matches).
__global__ void edge_att_kernel(const float* __restrict__ ent,
                                const float* __restrict__ weight,
                                const int* __restrict__ eidx,
                                const int* __restrict__ etype,
                                float* __restrict__ att,
                                unsigned int* __restrict__ mmax,
                                int E) {
  int e    = (int)(((long long)blockIdx.x * blockDim.x + threadIdx.x) >> 5);
  int lane = threadIdx.x & 31;
  if (e >= E) return;                      // wave-uniform branch
  int head = eidx[e];
  int tail = eidx[E + e];
  int r    = etype[e] - 1;

  const float4* h4 = (const float4*)(ent + (long long)head * D128);
  const float4* t4 = (const float4*)(ent + (long long)tail * D128);
  const float4* w4 = (const float4*)(weight + (long long)r * D128);
  float4 h = h4[lane], t = t4[lane], w = w4[lane];

  float hx = h.x * w.x, hy = h.y * w.y, hz = h.z * w.z, hw = h.w * w.w;
  float tx = t.x * w.x, ty = t.y * w.y, tz = t.z * w.z, tw = t.w * w.w;
  float hn2 = hx * hx + hy * hy + hz * hz + hw * hw;
  float tn2 = tx * tx + ty * ty + tz * tz + tw * tw;

  for (int off = 16; off; off >>= 1) {     // wave32 butterfly reduce
    hn2 += __shfl_xor(hn2, off, 32);
    tn2 += __shfl_xor(tn2, off, 32);
  }
  if (lane == 0) {
    float a = hn2 * tn2;
    att[e] = a;
    atomicMax(mmax + head, __float_as_uint(a));
  }
}

// ------------------------------------------------- pass 2: exp + segment sum
__global__ void edge_expsum_kernel(const int* __restrict__ eidx,
                                   const unsigned int* __restrict__ mmax,
                                   float* __restrict__ att,
                                   float* __restrict__ ssum, int E) {
  int e = blockIdx.x * blockDim.x + threadIdx.x;
  if (e >= E) return;
  int head = eidx[e];
  float m  = __uint_as_float(mmax[head]);
  float ev = __expf(att[e] - m);
  att[e] = ev;                             // overwrite att with exp value
  atomicAdd(ssum + head, ev);
}

// -------------------------------------- pass 3: weighted scatter into entity_agg
__global__ void edge_scatter_kernel(const float* __restrict__ ent,
                                    const float* __restrict__ weight,
                                    const int* __restrict__ eidx,
                                    const int* __restrict__ etype,
                                    const float* __restrict__ att,
                                    const float* __restrict__ ssum,
                                    float* __restrict__ entity_agg, int E) {
  int e    = (int)(((long long)blockIdx.x * blockDim.x + threadIdx.x) >> 5);
  int lane = threadIdx.x & 31;
  if (e >= E) return;
  int head = eidx[e];
  int tail = eidx[E + e];
  int r    = etype[e] - 1;
  float wgt = att[e] / ssum[head];

  const float4* t4 = (const float4*)(ent + (long long)tail * D128);
  const float4* w4 = (const float4*)(weight + (long long)r * D128);
  float4 t = t4[lane], w = w4[lane];

  float* dst = entity_agg + (long long)head * D128 + lane * 4;
  atomicAdd(dst + 0, wgt * t.x * w.x);
  atomicAdd(dst + 1, wgt * t.y * w.y);
  atomicAdd(dst + 2, wgt * t.z * w.z);
  atomicAdd(dst + 3, wgt * t.w * w.w);
}

// -------------------------------------------- COO interact_mat @ entity_emb
__global__ void user_scatter_kernel(const float* __restrict__ ent,
                                    const int* __restrict__ rows,
                                    const int* __restrict__ cols,
                                    const float* __restrict__ vals,
                                    float* __restrict__ user_agg, int nnz) {
  int i    = (int)(((long long)blockIdx.x * blockDim.x + threadIdx.x) >> 5);
  int lane = threadIdx.x & 31;
  if (i >= nnz) return;
  int row = rows[i];
  int col = cols[i];
  float v = vals[i];

  const float4* c4 = (const float4*)(ent + (long long)col * D128);
  float4 c = c4[lane];
  float* dst = user_agg + (long long)row * D128 + lane * 4;
  atomicAdd(dst + 0, v * c.x);
  atomicAdd(dst + 1, v * c.y);
  atomicAdd(dst + 2, v * c.z);
  atomicAdd(dst + 3, v * c.w);
}

// ---------------------------------------------------------------------------
// One wave per 16-user tile:
//   S = user_emb_tile[16x128] @ weight^T[128x32]     (V_WMMA_F32_16X16X4_F32)
//   score = softmax(S) rowwise (32 wide, via LDS)
//   P = score[16x32] @ weight[32x128]                (V_WMMA_F32_16X16X4_F32)
//   user_agg = user_agg * (1 + P)                    (in place on d_out)
//
// Out-of-range rows are handled by CLAMPING the load row (always a valid
// address -> unconditional global_load_b64, no exec-mask churn) and guarding
// only the stores. Garbage rows never reach memory.
// ---------------------------------------------------------------------------
__global__ __launch_bounds__(32) void user_gemm_kernel(
    const float* __restrict__ uemb, const float* __restrict__ weight,
    float* __restrict__ uagg, int n_users) {
  __shared__ float S[16][33];              // +1 pad vs bank conflicts

  int lane  = threadIdx.x;
  int m     = lane & 15;                   // row (A) / col (B,C/D) within tile
  int hi    = lane >> 4;                   // 0: K=0,1   1: K=2,3
  int koff  = hi << 1;
  int rbase = hi << 3;                     // C/D rows: 0..7 or 8..15
  int row0  = blockIdx.x * 16;

  // ---- GEMM1: S = A (16x128) x W^T (128x32), two 16x16 N-tiles ----
  int arow = row0 + m;
  int lrow = arow < n_users ? arow : (n_users - 1);   // clamped load row
  const float* ap  = uemb + (long long)lrow * D128 + koff;
  const float* b0p = weight + (long long)m * D128 + koff;         // rels 0..15
  const float* b1p = weight + (long long)(16 + m) * D128 + koff;  // rels 16..31

  v8f c0 = {}; v8f c1 = {};
  for (int k0 = 0; k0 < 128; k0 += 4) {
    v2f a, b0, b1;
    a.x  = ap[k0];   a.y  = ap[k0 + 1];     // unconditional -> global_load_b64
    b0.x = b0p[k0];  b0.y = b0p[k0 + 1];
    b1.x = b1p[k0];  b1.y = b1p[k0 + 1];
    c0 = __builtin_amdgcn_wmma_f32_16x16x4_f32(false, a, false, b0,
                                               (short)0, c0, false, false);
    c1 = __builtin_amdgcn_wmma_f32_16x16x4_f32(false, a, false, b1,
                                               (short)0, c1, false, false);
  }

  // spill S tile to LDS: lane holds col m, rows rbase..rbase+7
  for (int r = 0; r < 8; ++r) {
    S[rbase + r][m]      = c0[r];
    S[rbase + r][16 + m] = c1[r];
  }
  __syncthreads();

  // ---- rowwise softmax over 32 relations: lanes 0..15 own one row each ----
  if (lane < 16) {
    float mx = -1e30f;
    for (int j = 0; j < 32; ++j) mx = fmaxf(mx, S[lane][j]);
    float ex[32];
    float sum = 0.0f;
    for (int j = 0; j < 32; ++j) { ex[j] = __expf(S[lane][j] - mx); sum += ex[j]; }
    float inv = 1.0f / sum;
    for (int j = 0; j < 32; ++j) S[lane][j] = ex[j] * inv;
  }
  __syncthreads();

  // ---- GEMM2: P = score (16x32) x W (32x128), 8 N-tiles ----
  for (int nt = 0; nt < 8; ++nt) {
    v8f cp = {};
    const float* bw = weight + nt * 16 + m;  // B[k][n] = weight[k*128 + nt*16+n]
    for (int k0 = 0; k0 < 32; k0 += 4) {
      v2f a, b;
      a.x = S[m][k0 + koff];
      a.y = S[m][k0 + koff + 1];
      b.x = bw[(long long)(k0 + koff) * D128];
      b.y = bw[(long long)(k0 + koff + 1) * D128];
      cp = __builtin_amdgcn_wmma_f32_16x16x4_f32(false, a, false, b,
                                                 (short)0, cp, false, false);
    }
    for (int r = 0; r < 8; ++r) {
      int row = row0 + rbase + r;
      if (row < n_users) {
        long long idx = (long long)row * D128 + nt * 16 + m;
        float u = uagg[idx];
        uagg[idx] = u + cp[r] * u;           // user_agg * (1 + P)
      }
    }
  }
}

// ---------------------------------------------------------------------------
extern "C" void kernel_launch(void* const* d_in, const int* in_sizes, int n_in,
                              void* d_out, int out_size, void* d_ws, size_t ws_size,
                              hipStream_t stream) {
  const float* ent    = (const float*)d_in[0];
  const float* uemb   = (const float*)d_in[1];
  const float* weight = (const float*)d_in[2];
  const int*   eidx   = (const int*)d_in[3];
  const int*   etype  = (const int*)d_in[4];
  const int*   irows  = (const int*)d_in[5];
  const int*   icols  = (const int*)d_in[6];
  const float* ivals  = (const float*)d_in[7];

  int n_ent   = in_sizes[0] / D128;
  int n_users = in_sizes[1] / D128;
  int E       = in_sizes[3] / 2;
  int nnz     = in_sizes[5];

  float* entity_agg = (float*)d_out;
  float* user_agg   = entity_agg + (long long)n_ent * D128;

  // workspace layout (floats): [mmax: n_ent][ssum: n_ent][att: E]
  float*        ws   = (float*)d_ws;
  unsigned int* mmax = (unsigned int*)ws;
  float*        ssum = ws + n_ent;
  float*        att  = ws + 2 * (long long)n_ent;

  // init: zero outputs + segment max/sum (att is fully overwritten by pass 1)
  zero_kernel<<<2048, 256, 0, stream>>>((float*)d_out, (long long)out_size);
  zero_kernel<<<512, 256, 0, stream>>>(ws, 2LL * n_ent);

  // pass 1: attention logits + segment max (wave per edge)
  edge_att_kernel<<<(E + 7) / 8, 256, 0, stream>>>(ent, weight, eidx, etype,
                                                   att, mmax, E);
  // pass 2: exp + segment sum (thread per edge)
  edge_expsum_kernel<<<(E + 255) / 256, 256, 0, stream>>>(eidx, mmax, att, ssum, E);
  // pass 3: softmax-weighted scatter into entity_agg (wave per edge)
  edge_scatter_kernel<<<(E + 7) / 8, 256, 0, stream>>>(ent, weight, eidx, etype,
                                                       att, ssum, entity_agg, E);
  // sparse interact_mat @ entity_emb (wave per nnz)
  user_scatter_kernel<<<(nnz + 7) / 8, 256, 0, stream>>>(ent, irows, icols,
                                                         ivals, user_agg, nnz);
  // WMMA: score = softmax(U W^T);  user_agg *= (1 + score W)
  user_gemm_kernel<<<(n_users + 15) / 16, 32, 0, stream>>>(uemb, weight,
                                                           user_agg, n_users);
}